// Capsule_81844896792661
// MI455X (gfx1250) — compile-verified
//
#include <hip/hip_runtime.h>
#include <hip/hip_bf16.h>

#define BATCH        32
#define N_TOK        2048
#define D_IN         128
#define NUM_CAPSULE  32
#define DIM_CAPSULE  16
#define W_COLS       (NUM_CAPSULE * DIM_CAPSULE)   // 512
#define EPS          1e-7f

typedef __attribute__((ext_vector_type(2))) float v2f;
typedef __attribute__((ext_vector_type(8))) float v8f;

// ---------------------------------------------------------------------------
// Pass 0a: partial column sums of inputs over n (deterministic two-stage).
// s_part[(b*16+chunk)*128 + k] = sum over 128 rows of inputs[b, n, k]
// ---------------------------------------------------------------------------
__global__ void k_colsum_part(const float* __restrict__ in, float* __restrict__ s_part) {
    int b = blockIdx.x;          // 32
    int chunk = blockIdx.y;      // 16 chunks of 128 rows
    int k = threadIdx.x;         // 128
    const float* p = in + ((size_t)b * N_TOK + (size_t)chunk * 128) * D_IN + k;
    float acc = 0.f;
    #pragma unroll 8
    for (int n = 0; n < 128; ++n) acc += p[(size_t)n * D_IN];
    s_part[((size_t)b * 16 + chunk) * D_IN + k] = acc;
}

// Pass 0b: reduce partials, apply uniform coupling 1/32.
// s_buf[b*NUM_CAPSULE*D_IN + k]  (i.e. stored at capsule slot 0)
__global__ void k_colsum_reduce(const float* __restrict__ s_part, float* __restrict__ s_buf) {
    int b = blockIdx.x;          // 32
    int k = threadIdx.x;         // 128
    float acc = 0.f;
    #pragma unroll
    for (int ch = 0; ch < 16; ++ch) acc += s_part[((size_t)b * 16 + ch) * D_IN + k];
    s_buf[(size_t)b * NUM_CAPSULE * D_IN + k] = acc * (1.0f / 32.0f);
}

// ---------------------------------------------------------------------------
// outputs[b,c,:] = squash( s[b,c,:] @ W[:, c*16 : c*16+16] )   (squash = x/sqrt(|x|^2+eps))
// v[b,c,k]      = sum_d outputs[b,c,d] * W[k, c*16+d]          (optional)
// cstride: 0 -> read s at capsule slot 0 (round 0, uniform c), 1 -> per-capsule s.
// ---------------------------------------------------------------------------
__global__ void k_outv(const float* __restrict__ s_buf, const float* __restrict__ W,
                       float* __restrict__ v_buf, float* __restrict__ out, int cstride) {
    int c = blockIdx.x;          // 32
    int b = blockIdx.y;          // 32
    int t = threadIdx.x;         // 128
    __shared__ float sh_out[DIM_CAPSULE];
    __shared__ float sh_inv;

    const float* s = s_buf + ((size_t)b * NUM_CAPSULE + (size_t)c * cstride) * D_IN;
    if (t < DIM_CAPSULE) {
        float raw = 0.f;
        #pragma unroll 4
        for (int k = 0; k < D_IN; ++k)
            raw += s[k] * W[(size_t)k * W_COLS + c * DIM_CAPSULE + t];
        sh_out[t] = raw;
    }
    __syncthreads();
    if (t == 0) {
        float ss = 0.f;
        #pragma unroll
        for (int d = 0; d < DIM_CAPSULE; ++d) ss += sh_out[d] * sh_out[d];
        sh_inv = 1.0f / sqrtf(ss + EPS);
    }
    __syncthreads();
    float inv = sh_inv;
    if (out != nullptr && t < DIM_CAPSULE)
        out[((size_t)b * NUM_CAPSULE + c) * DIM_CAPSULE + t] = sh_out[t] * inv;
    if (v_buf != nullptr) {
        const float* wrow = W + (size_t)t * W_COLS + c * DIM_CAPSULE;  // 16 contiguous floats
        float acc = 0.f;
        #pragma unroll
        for (int d = 0; d < DIM_CAPSULE; ++d) acc += (sh_out[d] * inv) * wrow[d];
        v_buf[((size_t)b * NUM_CAPSULE + c) * D_IN + t] = acc;
    }
}

// ---------------------------------------------------------------------------
// Logits via f32 WMMA:  L[b,n,c] = sum_k inputs[b,n,k] * v[b,c,k]
// Per batch: A = inputs[b] (2048x128), B[k][c] = v[b][c][k] (128x32).
// Block = 128 thr = 4 waves: wave covers (n-subtile 16, c-half 16). K-loop: 32 WMMAs.
// b_buf layout: [b][n][c]  (contiguous over c for the softmax kernel)
// ---------------------------------------------------------------------------
__global__ void k_logits(const float* __restrict__ in, const float* __restrict__ v_buf,
                         float* __restrict__ b_buf) {
    int ntile = blockIdx.x;           // 64 tiles of 32 rows
    int b     = blockIdx.y;           // 32
    int wave  = threadIdx.x >> 5;     // 0..3
    int lane  = threadIdx.x & 31;
    int lm    = lane & 15;
    int half  = (lane >> 4) << 1;     // 0 or 2 (K offset within a 4-step)
    int n0    = ntile * 32 + (wave & 1) * 16;
    int c0    = (wave >> 1) * 16;

    // A fragment source: row n0+lm of inputs[b]; per step lane holds K = kb+half, kb+half+1
    const float* arow = in    + ((size_t)b * N_TOK + (n0 + lm)) * D_IN;
    // B fragment source: column c0+lm of B, i.e. v[b][c0+lm][*] (contiguous in k)
    const float* brow = v_buf + ((size_t)b * NUM_CAPSULE + (c0 + lm)) * D_IN;

    v8f acc = {};
    #pragma unroll 4
    for (int kb = 0; kb < D_IN; kb += 4) {
        v2f a  = *(const v2f*)(arow + kb + half);
        v2f bb = *(const v2f*)(brow + kb + half);
        acc = __builtin_amdgcn_wmma_f32_16x16x4_f32(false, a, false, bb,
                                                    (short)0, acc, false, false);
    }
    // C/D layout: VGPR r -> (M = r (+8 for lanes>=16), N = lane&15)
    int cc    = c0 + lm;
    int nbase = n0 + ((lane >> 4) << 3);
    float* dst = b_buf + ((size_t)b * N_TOK + nbase) * NUM_CAPSULE + cc;
    #pragma unroll
    for (int r = 0; r < 8; ++r) dst[(size_t)r * NUM_CAPSULE] = acc[r];
}

// ---------------------------------------------------------------------------
// Softmax over the capsule axis for each (b,n). Reads [b][n][c], writes [b][c][n]
// (transposed so that k_s gets contiguous A-fragment loads).
// ---------------------------------------------------------------------------
__global__ void k_softmax(const float* __restrict__ b_buf, float* __restrict__ c_buf) {
    int id = blockIdx.x * blockDim.x + threadIdx.x;   // 0 .. 65535  (= b*2048+n)
    int b  = id >> 11;
    int n  = id & (N_TOK - 1);
    const float* p = b_buf + (size_t)id * NUM_CAPSULE;
    float vals[NUM_CAPSULE];
    float m = -INFINITY;
    #pragma unroll
    for (int c = 0; c < NUM_CAPSULE; ++c) { vals[c] = p[c]; m = fmaxf(m, vals[c]); }
    float sum = 0.f;
    #pragma unroll
    for (int c = 0; c < NUM_CAPSULE; ++c) { vals[c] = __expf(vals[c] - m); sum += vals[c]; }
    float inv = 1.0f / sum;
    #pragma unroll
    for (int c = 0; c < NUM_CAPSULE; ++c)
        c_buf[((size_t)b * NUM_CAPSULE + c) * N_TOK + n] = vals[c] * inv;   // coalesced over n
}

// ---------------------------------------------------------------------------
// s via f32 WMMA:  s[b,c,kf] = sum_n c_buf[b,c,n] * inputs[b,n,kf]
// Per batch: A = c (32x2048), B = inputs[b] (2048x128). K-loop over n: 512 WMMAs.
// Block = 128 thr = 4 waves: wave = (c-tile 0/1, feature-tile). Grid.x = 4 feature groups.
// ---------------------------------------------------------------------------
__global__ void k_s(const float* __restrict__ c_buf, const float* __restrict__ in,
                    float* __restrict__ s_buf) {
    int ntg  = blockIdx.x;            // 4 groups of 2 feature tiles
    int b    = blockIdx.y;            // 32
    int wave = threadIdx.x >> 5;      // 0..3
    int lane = threadIdx.x & 31;
    int lm   = lane & 15;
    int half = (lane >> 4) << 1;
    int m0   = (wave & 1) * 16;                 // capsule tile base
    int n0   = (ntg * 2 + (wave >> 1)) * 16;    // feature (k) tile base

    const float* arow = c_buf + ((size_t)b * NUM_CAPSULE + (m0 + lm)) * N_TOK; // contiguous in n
    const float* bcol = in    + (size_t)b * N_TOK * D_IN + (n0 + lm);          // stride D_IN over n

    v8f acc = {};
    #pragma unroll 2
    for (int kb = 0; kb < N_TOK; kb += 4) {
        v2f a = *(const v2f*)(arow + kb + half);
        v2f bb;
        bb.x = bcol[(size_t)(kb + half)     * D_IN];
        bb.y = bcol[(size_t)(kb + half + 1) * D_IN];
        acc = __builtin_amdgcn_wmma_f32_16x16x4_f32(false, a, false, bb,
                                                    (short)0, acc, false, false);
    }
    int kf    = n0 + lm;
    int cbase = m0 + ((lane >> 4) << 3);
    float* dst = s_buf + ((size_t)b * NUM_CAPSULE + cbase) * D_IN + kf;
    #pragma unroll
    for (int r = 0; r < 8; ++r) dst[(size_t)r * D_IN] = acc[r];
}

// ---------------------------------------------------------------------------
extern "C" void kernel_launch(void* const* d_in, const int* in_sizes, int n_in,
                              void* d_out, int out_size, void* d_ws, size_t ws_size,
                              hipStream_t stream) {
    const float* inputs = (const float*)d_in[0];   // (32, 2048, 128)
    const float* W      = (const float*)d_in[1];   // (1, 128, 512)
    float* out          = (float*)d_out;           // (32, 32, 16)

    char* ws = (char*)d_ws;
    const size_t S_BYTES = (size_t)BATCH * NUM_CAPSULE * D_IN * sizeof(float);   // 512 KB
    const size_t B_BYTES = (size_t)BATCH * N_TOK * NUM_CAPSULE * sizeof(float);  // 8 MB
    float* s_buf  = (float*)(ws);
    float* v_buf  = (float*)(ws + S_BYTES);
    float* b_buf  = (float*)(ws + 2 * S_BYTES);
    float* c_buf  = (float*)(ws + 2 * S_BYTES + B_BYTES);
    float* s_part = (float*)(ws + 2 * S_BYTES + 2 * B_BYTES);                    // 256 KB

    // Round 0: uniform coupling -> s is capsule-independent column sum / 32.
    k_colsum_part  <<<dim3(BATCH, 16), 128, 0, stream>>>(inputs, s_part);
    k_colsum_reduce<<<BATCH,           128, 0, stream>>>(s_part, s_buf);
    k_outv<<<dim3(NUM_CAPSULE, BATCH), 128, 0, stream>>>(s_buf, W, v_buf, nullptr, /*cstride=*/0);

    // Rounds 1 and 2.
    for (int it = 0; it < 2; ++it) {
        const bool last = (it == 1);
        k_logits <<<dim3(64, BATCH), 128, 0, stream>>>(inputs, v_buf, b_buf);
        k_softmax<<<(BATCH * N_TOK) / 256, 256, 0, stream>>>(b_buf, c_buf);
        k_s      <<<dim3(4, BATCH), 128, 0, stream>>>(c_buf, inputs, s_buf);
        k_outv   <<<dim3(NUM_CAPSULE, BATCH), 128, 0, stream>>>(
            s_buf, W, last ? nullptr : v_buf, last ? out : nullptr, /*cstride=*/1);
    }
}